// GCN_67568425500960
// MI455X (gfx1250) — compile-verified
//
#include <hip/hip_runtime.h>
#include <hip/hip_bf16.h>

#define NNODE 3072
#define FIN   1546
#define D0    256
#define HDIM  128
#define BPAIR 16384

typedef __attribute__((ext_vector_type(2))) float v2f;
typedef __attribute__((ext_vector_type(8))) float v8f;

// 8 WMMAs: 2 A fragments (M strips) x 4 B fragments (N strips)
__device__ __forceinline__ void wmma8(v8f (&acc)[2][4], v2f a0, v2f a1,
                                      v2f b0, v2f b1, v2f b2, v2f b3)
{
  acc[0][0] = __builtin_amdgcn_wmma_f32_16x16x4_f32(false, a0, false, b0, (short)0, acc[0][0], false, false);
  acc[0][1] = __builtin_amdgcn_wmma_f32_16x16x4_f32(false, a0, false, b1, (short)0, acc[0][1], false, false);
  acc[0][2] = __builtin_amdgcn_wmma_f32_16x16x4_f32(false, a0, false, b2, (short)0, acc[0][2], false, false);
  acc[0][3] = __builtin_amdgcn_wmma_f32_16x16x4_f32(false, a0, false, b3, (short)0, acc[0][3], false, false);
  acc[1][0] = __builtin_amdgcn_wmma_f32_16x16x4_f32(false, a1, false, b0, (short)0, acc[1][0], false, false);
  acc[1][1] = __builtin_amdgcn_wmma_f32_16x16x4_f32(false, a1, false, b1, (short)0, acc[1][1], false, false);
  acc[1][2] = __builtin_amdgcn_wmma_f32_16x16x4_f32(false, a1, false, b2, (short)0, acc[1][2], false, false);
  acc[1][3] = __builtin_amdgcn_wmma_f32_16x16x4_f32(false, a1, false, b3, (short)0, acc[1][3], false, false);
}

// ---------------------------------------------------------------------------
// fp32 WMMA GEMM: C = act(A @ B + bias). Block = 128 threads = 4 wave32s.
// Each wave computes a 32(M) x 64(N) tile (8 accumulators); block tile is
// 128(M) x 64(N). Grid must be (N/64, M/128). Hot loop is branch-free with
// b64 A loads; the only guarded (divergent) code is a single tail step when
// K % 4 != 0 (fc1 only, K = 1546).
// GATHER: A rows indirected via left/right (pair-concat, K fixed = 512).
// ---------------------------------------------------------------------------
template <bool RELU, bool BIAS, bool GATHER>
__global__ __launch_bounds__(128) void gemm_wmma_f32(
    const float* __restrict__ A, int lda,
    const float* __restrict__ Bm, int ldb,
    const float* __restrict__ bias,
    float* __restrict__ C, int ldc,
    int K,
    const int* __restrict__ gl, const int* __restrict__ gr)
{
  const int lane = threadIdx.x & 31;
  const int wave = threadIdx.x >> 5;
  const int half = lane >> 4;     // ISA: VGPR v holds K = v + 2*half
  const int l16  = lane & 15;

  const int m0  = blockIdx.y * 128 + wave * 32;
  const int n0  = blockIdx.x * 64;
  const int am0 = m0 + l16;        // A row, M strip 0
  const int am1 = m0 + 16 + l16;   // A row, M strip 1

  v8f acc[2][4] = {};

  const float* bp = Bm + (size_t)(2 * half) * ldb + n0 + l16;
  const size_t bstep = (size_t)4 * ldb;

  if (GATHER) {
    // concat(emb[left], emb[right]): two dense 256-wide halves, 64 steps each
    const float* ap0 = A + (size_t)gl[am0] * 256 + 2 * half;
    const float* ap1 = A + (size_t)gl[am1] * 256 + 2 * half;
    #pragma unroll 4
    for (int k = 0; k < 256; k += 4) {
      v2f a0 = *(const v2f*)ap0;
      v2f a1 = *(const v2f*)ap1;
      v2f b0, b1, b2, b3;
      b0.x = bp[0];  b0.y = bp[ldb + 0];
      b1.x = bp[16]; b1.y = bp[ldb + 16];
      b2.x = bp[32]; b2.y = bp[ldb + 32];
      b3.x = bp[48]; b3.y = bp[ldb + 48];
      wmma8(acc, a0, a1, b0, b1, b2, b3);
      ap0 += 4; ap1 += 4; bp += bstep;
    }
    ap0 = A + (size_t)gr[am0] * 256 + 2 * half;
    ap1 = A + (size_t)gr[am1] * 256 + 2 * half;
    #pragma unroll 4
    for (int k = 0; k < 256; k += 4) {
      v2f a0 = *(const v2f*)ap0;
      v2f a1 = *(const v2f*)ap1;
      v2f b0, b1, b2, b3;
      b0.x = bp[0];  b0.y = bp[ldb + 0];
      b1.x = bp[16]; b1.y = bp[ldb + 16];
      b2.x = bp[32]; b2.y = bp[ldb + 32];
      b3.x = bp[48]; b3.y = bp[ldb + 48];
      wmma8(acc, a0, a1, b0, b1, b2, b3);
      ap0 += 4; ap1 += 4; bp += bstep;
    }
  } else {
    const float* ap0 = A + (size_t)am0 * lda + 2 * half;
    const float* ap1 = A + (size_t)am1 * lda + 2 * half;
    const int kmain = K & ~3;
    #pragma unroll 2
    for (int k = 0; k < kmain; k += 4) {
      v2f a0 = *(const v2f*)ap0;          // 8B-aligned: lda even, 2*half even
      v2f a1 = *(const v2f*)ap1;
      v2f b0, b1, b2, b3;
      b0.x = bp[0];  b0.y = bp[ldb + 0];
      b1.x = bp[16]; b1.y = bp[ldb + 16];
      b2.x = bp[32]; b2.y = bp[ldb + 32];
      b3.x = bp[48]; b3.y = bp[ldb + 48];
      wmma8(acc, a0, a1, b0, b1, b2, b3);
      ap0 += 4; ap1 += 4; bp += bstep;
    }
    if (kmain < K) {                      // single guarded tail (K % 4 != 0)
      const int ka = kmain + 2 * half;
      const bool g0 = (ka < K), g1 = (ka + 1 < K);
      v2f a0, a1, b0, b1, b2, b3;
      a0.x = g0 ? ap0[0] : 0.0f;  a0.y = g1 ? ap0[1] : 0.0f;
      a1.x = g0 ? ap1[0] : 0.0f;  a1.y = g1 ? ap1[1] : 0.0f;
      b0.x = g0 ? bp[0]  : 0.0f;  b0.y = g1 ? bp[ldb + 0]  : 0.0f;
      b1.x = g0 ? bp[16] : 0.0f;  b1.y = g1 ? bp[ldb + 16] : 0.0f;
      b2.x = g0 ? bp[32] : 0.0f;  b2.y = g1 ? bp[ldb + 32] : 0.0f;
      b3.x = g0 ? bp[48] : 0.0f;  b3.y = g1 ? bp[ldb + 48] : 0.0f;
      wmma8(acc, a0, a1, b0, b1, b2, b3);
    }
  }

  // C/D layout: VGPR v -> row m0 + 16*s + v + 8*half, col n0 + 16*j + l16
  #pragma unroll
  for (int s = 0; s < 2; ++s) {
    #pragma unroll
    for (int v = 0; v < 8; ++v) {
      const int row = m0 + 16 * s + v + 8 * half;
      #pragma unroll
      for (int j = 0; j < 4; ++j) {
        const int col = n0 + 16 * j + l16;
        float x = acc[s][j][v];
        if (BIAS) x += bias[col];
        if (RELU) x = fmaxf(x, 0.0f);
        C[(size_t)row * ldc + col] = x;
      }
    }
  }
}

// ------------------------- row degree -> rsqrt ------------------------------
__global__ void deg_rsqrt(const float* __restrict__ A, float* __restrict__ d)
{
  const int gw   = (blockIdx.x * blockDim.x + threadIdx.x) >> 5;
  const int lane = threadIdx.x & 31;
  if (gw >= NNODE) return;
  const float* row = A + (size_t)gw * NNODE;
  float s = 0.0f;
  for (int j = lane; j < NNODE; j += 32) s += row[j];
  #pragma unroll
  for (int off = 16; off > 0; off >>= 1) s += __shfl_xor(s, off, 32);
  if (lane == 0) d[gw] = (s > 0.0f) ? rsqrtf(s) : 0.0f;
}

// branch degree: deg = 1 (self loop) + sum_j (mask > thr ? adj : 0); always > 0
__global__ void branch_deg_rsqrt(const float* __restrict__ mask,
                                 const float* __restrict__ adj,
                                 const float* __restrict__ thrp,
                                 float* __restrict__ d)
{
  const int gw   = (blockIdx.x * blockDim.x + threadIdx.x) >> 5;
  const int lane = threadIdx.x & 31;
  if (gw >= NNODE) return;
  const float thr = thrp[0];
  const size_t base = (size_t)gw * NNODE;
  float s = 0.0f;
  for (int j = lane; j < NNODE; j += 32)
    s += (mask[base + j] > thr) ? adj[base + j] : 0.0f;
  #pragma unroll
  for (int off = 16; off > 0; off >>= 1) s += __shfl_xor(s, off, 32);
  if (lane == 0) d[gw] = rsqrtf(s + 1.0f);
}

// An = D^-1/2 A D^-1/2 (no self loops)
__global__ void symnorm(const float* __restrict__ adj,
                        const float* __restrict__ d,
                        float* __restrict__ An)
{
  const int idx = blockIdx.x * blockDim.x + threadIdx.x;
  if (idx >= NNODE * NNODE) return;
  const int i = idx / NNODE;
  const int j = idx - i * NNODE;
  An[idx] = adj[idx] * d[i] * d[j];
}

// AnB = D^-1/2 (mask-thresholded adj + I) D^-1/2  (fused, no Wadj scratch)
__global__ void branch_norm(const float* __restrict__ mask,
                            const float* __restrict__ adj,
                            const float* __restrict__ thrp,
                            const float* __restrict__ d,
                            float* __restrict__ An)
{
  const int idx = blockIdx.x * blockDim.x + threadIdx.x;
  if (idx >= NNODE * NNODE) return;
  const float thr = thrp[0];
  const int i = idx / NNODE;
  const int j = idx - i * NNODE;
  float w = (mask[idx] > thr) ? adj[idx] : 0.0f;
  if (i == j) w += 1.0f;
  An[idx] = w * d[i] * d[j];
}

// 3x3 conv, rhs_dilation=2, padding=2, 3 input channels -> 1 output channel
__global__ void conv_dil2(const float* __restrict__ emb,
                          const float* __restrict__ kw,
                          const float* __restrict__ kb,
                          float* __restrict__ out)
{
  const int idx = blockIdx.x * blockDim.x + threadIdx.x;
  if (idx >= NNODE * 256) return;
  const int n = idx >> 8;
  const int c = idx & 255;
  float acc = kb[0];
  #pragma unroll
  for (int ch = 0; ch < 3; ++ch) {
    #pragma unroll
    for (int kh = 0; kh < 3; ++kh) {
      const int ih = n + 2 * kh - 2;
      if (ih < 0 || ih >= NNODE) continue;
      #pragma unroll
      for (int kwi = 0; kwi < 3; ++kwi) {
        const int iw = c + 2 * kwi - 2;
        if (iw < 0 || iw >= 256) continue;
        acc += emb[((size_t)ch * NNODE + ih) * 256 + iw] * kw[ch * 9 + kh * 3 + kwi];
      }
    }
  }
  out[idx] = acc;
}

// logits = h @ W_b + b_b    (16384x64 @ 64x2)
__global__ void logits_k(const float* __restrict__ h,
                         const float* __restrict__ Wb,
                         const float* __restrict__ bb,
                         float* __restrict__ out)
{
  const int i = blockIdx.x * blockDim.x + threadIdx.x;
  if (i >= BPAIR * 2) return;
  const int b = i >> 1;
  const int c = i & 1;
  float acc = bb[c];
  #pragma unroll
  for (int k = 0; k < 64; ++k) acc += h[b * 64 + k] * Wb[k * 2 + c];
  out[i] = acc;
}

// ---------------------------------------------------------------------------
extern "C" void kernel_launch(void* const* d_in, const int* in_sizes, int n_in,
                              void* d_out, int out_size, void* d_ws, size_t ws_size,
                              hipStream_t stream)
{
  const int*   left  = (const int*)d_in[0];
  const int*   right = (const int*)d_in[1];
  const float* X     = (const float*)d_in[2];
  const float* adj   = (const float*)d_in[3];
  const float* thr   = (const float*)d_in[4];
  const float* Wfc1  = (const float*)d_in[5];
  const float* bfc1  = (const float*)d_in[6];
  const float* Wg[6] = {(const float*)d_in[7],  (const float*)d_in[9],
                        (const float*)d_in[11], (const float*)d_in[13],
                        (const float*)d_in[15], (const float*)d_in[17]};
  const float* bg[6] = {(const float*)d_in[8],  (const float*)d_in[10],
                        (const float*)d_in[12], (const float*)d_in[14],
                        (const float*)d_in[16], (const float*)d_in[18]};
  const float* cnnk  = (const float*)d_in[19];
  const float* cnnb  = (const float*)d_in[20];
  const float* Wa    = (const float*)d_in[21];
  const float* ba    = (const float*)d_in[22];
  const float* Wb    = (const float*)d_in[23];
  const float* bb    = (const float*)d_in[24];

  float* out     = (float*)d_out;
  float* logits  = out;                 // [16384, 2]
  float* emb_all = out + BPAIR * 2;     // [3072, 256]

  float* ws = (float*)d_ws;
  size_t off = 0;
  float* X0  = ws + off; off += (size_t)NNODE * D0;
  float* An  = ws + off; off += (size_t)NNODE * NNODE;
  float* A2  = ws + off; off += (size_t)NNODE * NNODE;
  float* AnB = ws + off; off += (size_t)NNODE * NNODE;
  float* dv  = ws + off; off += NNODE;
  float* Hb  = ws + off; off += (size_t)NNODE * HDIM;
  float* H2b = ws + off; off += (size_t)NNODE * HDIM;
  float* emb = ws + off; off += (size_t)3 * NNODE * 256;
  float* hm  = ws + off; off += (size_t)BPAIR * 64;
  if (ws_size < off * sizeof(float)) return;   // not enough scratch

  const dim3 blk(128);

  // 1) X0 = relu(X @ W_fc1 + b_fc1)            [3072,1546]x[1546,256]
  gemm_wmma_f32<true, true, false><<<dim3(D0 / 64, NNODE / 128), blk, 0, stream>>>(
      X, FIN, Wfc1, D0, bfc1, X0, D0, FIN, nullptr, nullptr);

  // 2) An = sym_norm(adj);  A2 = An @ An       [3072^3]
  deg_rsqrt<<<NNODE / 8, 256, 0, stream>>>(adj, dv);
  symnorm<<<(NNODE * NNODE) / 256, 256, 0, stream>>>(adj, dv, An);
  gemm_wmma_f32<false, false, false><<<dim3(NNODE / 64, NNODE / 128), blk, 0, stream>>>(
      An, NNODE, An, NNODE, nullptr, A2, NNODE, NNODE, nullptr, nullptr);

  // 3) branches: ch0 mask=adj (Wg5/Wg6), ch1 mask=An (Wg3/Wg4), ch2 mask=A2 (Wg1/Wg2)
  const float* masks[3] = { adj, An, A2 };
  for (int ch = 0; ch < 3; ++ch) {
    const float* W1 = Wg[4 - 2 * ch];  const float* b1 = bg[4 - 2 * ch];
    const float* W2 = Wg[5 - 2 * ch];  const float* b2 = bg[5 - 2 * ch];
    float* embc = emb + (size_t)ch * NNODE * 256;

    branch_deg_rsqrt<<<NNODE / 8, 256, 0, stream>>>(masks[ch], adj, thr, dv);
    branch_norm<<<(NNODE * NNODE) / 256, 256, 0, stream>>>(masks[ch], adj, thr, dv, AnB);

    // H = X0 @ W1
    gemm_wmma_f32<false, false, false><<<dim3(HDIM / 64, NNODE / 128), blk, 0, stream>>>(
        X0, D0, W1, HDIM, nullptr, Hb, HDIM, D0, nullptr, nullptr);
    // e1 = relu(AnB @ H + b1) -> embc[:, 0:128]
    gemm_wmma_f32<true, true, false><<<dim3(HDIM / 64, NNODE / 128), blk, 0, stream>>>(
        AnB, NNODE, Hb, HDIM, b1, embc, 256, NNODE, nullptr, nullptr);
    // H2 = e1 @ W2
    gemm_wmma_f32<false, false, false><<<dim3(HDIM / 64, NNODE / 128), blk, 0, stream>>>(
        embc, 256, W2, HDIM, nullptr, H2b, HDIM, HDIM, nullptr, nullptr);
    // e2 = relu(AnB @ H2 + b2) -> embc[:, 128:256]
    gemm_wmma_f32<true, true, false><<<dim3(HDIM / 64, NNODE / 128), blk, 0, stream>>>(
        AnB, NNODE, H2b, HDIM, b2, embc + 128, 256, NNODE, nullptr, nullptr);
  }

  // 4) dilated conv over the 3 stacked channels -> emb_all (written into d_out)
  conv_dil2<<<(NNODE * 256) / 256, 256, 0, stream>>>(emb, cnnk, cnnb, emb_all);

  // 5) pair MLP: h = relu(concat(emb[left], emb[right]) @ W_a + b_a)  (gather GEMM)
  gemm_wmma_f32<true, true, true><<<dim3(64 / 64, BPAIR / 128), blk, 0, stream>>>(
      emb_all, 256, Wa, 64, ba, hm, 64, 512, left, right);

  // 6) logits = h @ W_b + b_b
  logits_k<<<(BPAIR * 2 + 255) / 256, 256, 0, stream>>>(hm, Wb, bb, logits);
}